// InfoNCELoss_41558103556199
// MI455X (gfx1250) — compile-verified
//
#include <hip/hip_runtime.h>
#include <math.h>

#define D        256
#define TEMP_INV (1.0f / 0.07f)

typedef float v2f __attribute__((ext_vector_type(2)));
typedef float v8f __attribute__((ext_vector_type(8)));

__device__ __forceinline__ float wave_reduce_add(float v) {
#pragma unroll
    for (int o = 16; o >= 1; o >>= 1) v += __shfl_xor(v, o, 32);
    return v;
}
__device__ __forceinline__ float wave_reduce_max(float v) {
#pragma unroll
    for (int o = 16; o >= 1; o >>= 1) v = fmaxf(v, __shfl_xor(v, o, 32));
    return v;
}

// ---------------------------------------------------------------------------
// K1: per-row anchor/positive norms + pos_sim. One wave (32 lanes) per row.
// ---------------------------------------------------------------------------
__global__ void k_pos(const float* __restrict__ a, const float* __restrict__ p,
                      float* __restrict__ pos_sim, float* __restrict__ a_inv, int B) {
    int wave = threadIdx.x >> 5, lane = threadIdx.x & 31;
    int row  = blockIdx.x * 4 + wave;
    if (row >= B) return;
    const float4* ar = (const float4*)(a + (size_t)row * D);
    const float4* pr = (const float4*)(p + (size_t)row * D);
    float saa = 0.f, spp = 0.f, sap = 0.f;
#pragma unroll
    for (int i = 0; i < 2; i++) {
        float4 av = ar[lane + i * 32];
        float4 pv = pr[lane + i * 32];
        saa += av.x * av.x + av.y * av.y + av.z * av.z + av.w * av.w;
        spp += pv.x * pv.x + pv.y * pv.y + pv.z * pv.z + pv.w * pv.w;
        sap += av.x * pv.x + av.y * pv.y + av.z * pv.z + av.w * pv.w;
    }
    saa = wave_reduce_add(saa);
    spp = wave_reduce_add(spp);
    sap = wave_reduce_add(sap);
    if (lane == 0) {
        float inva = 1.0f / fmaxf(sqrtf(saa), 1e-12f);
        float invp = 1.0f / fmaxf(sqrtf(spp), 1e-12f);
        pos_sim[row] = sap * inva * invp * TEMP_INV;
        a_inv[row]   = inva;
    }
}

// ---------------------------------------------------------------------------
// K2: neg_sim[j] = <a_idx[j], n_j> / (|a| |n| T) via V_WMMA_F32_16X16X4_F32.
// Each wave handles 16 negatives; diagonal of the 16x16 accumulator holds
// the 16 dot products. K is chunked (64 cols) through padded LDS.
//
// A-operand (16x4 f32) layout per ISA 7.12.2: lanes 0-15 hold K=k0,k0+1;
// lanes 16-31 hold K=k0+2,k0+3 (row M = lane%16). B-operand (4x16) assumed
// mirrored: row N = lane%16 of the transposed tile with the same K split.
// ---------------------------------------------------------------------------
#define KC   64          // K-chunk columns
#define NW   4           // waves per block (128 threads)
#define TN   16          // negatives per wave
#define ROWP (KC + 4)    // padded row stride (floats): 68*4B -> 4-bank row skew

__global__ void __launch_bounds__(128) k_neg(
    const float* __restrict__ anchor, const float* __restrict__ neg,
    const int* __restrict__ idx, const float* __restrict__ a_inv,
    float* __restrict__ neg_sim, int N) {
    __shared__ __align__(16) float nlds[NW][TN][ROWP];
    __shared__ __align__(16) float alds[NW][TN][ROWP];
    (void)N;
    int wave  = threadIdx.x >> 5, lane = threadIdx.x & 31;
    int wbase = blockIdx.x * (NW * TN) + wave * TN;
    int m     = lane & 15;              // matrix row (A) / col (B) for this lane
    int aidx_m = idx[wbase + m];        // owning anchor of this lane's negative

    v8f   c   = {0.f, 0.f, 0.f, 0.f, 0.f, 0.f, 0.f, 0.f};
    float ssp = 0.0f;                   // per-lane partial sum-of-squares (half row)
    int   kodd = (lane < 16) ? 0 : 2;   // K sub-offset per lane half

#pragma unroll
    for (int ch = 0; ch < D / KC; ch++) {
        int k0 = ch * KC;
        // ---- stage 16 rows x KC cols (negatives + gathered anchors), coalesced f4
#pragma unroll
        for (int it = 0; it < (TN * KC / 4) / 32; it++) {   // 8 iterations
            int f  = it * 32 + lane;
            int r  = f >> 4;            // 16 float4 per row
            int c4 = f & 15;
            float4 nv = *(const float4*)(neg + (size_t)(wbase + r) * D + k0 + c4 * 4);
            *(float4*)&nlds[wave][r][c4 * 4] = nv;
            int ar = idx[wbase + r];
            float4 av = *(const float4*)(anchor + (size_t)ar * D + k0 + c4 * 4);
            *(float4*)&alds[wave][r][c4 * 4] = av;
        }
        __syncthreads();
        // ---- per-negative-row sum of squares (each lane: half of row m)
        {
            int co = (lane >> 4) * (KC / 2);
#pragma unroll
            for (int j = 0; j < KC / 8; j++) {
                float4 v = *(const float4*)&nlds[wave][m][co + j * 4];
                ssp += v.x * v.x + v.y * v.y + v.z * v.z + v.w * v.w;
            }
        }
        // ---- WMMA sweep over this K-chunk: C += A(16x4) * B(4x16)
#pragma unroll
        for (int s = 0; s < KC / 4; s++) {
            int ka = s * 4 + kodd;
            v2f av = *(const v2f*)&alds[wave][m][ka];
            v2f bv = *(const v2f*)&nlds[wave][m][ka];
            c = __builtin_amdgcn_wmma_f32_16x16x4_f32(
                /*neg_a=*/false, av, /*neg_b=*/false, bv,
                /*c_mod=*/(short)0, c, /*reuse_a=*/false, /*reuse_b=*/false);
        }
        __syncthreads();
    }

    // full sum-of-squares for row m (combine the two half-row partials)
    float ss = ssp + __shfl_xor(ssp, 16, 32);

    // diagonal extraction from the 16x16 f32 C layout:
    //   lanes 0-7  hold diag j=lane   in accumulator element lane
    //   lanes 24-31 hold diag j=lane-16 in accumulator element lane-24
    bool active = (lane < 8) || (lane >= 24);
    int  sel    = (lane < 8) ? lane : (lane - 24);
    float dv = c[0];
#pragma unroll
    for (int i = 1; i < 8; i++)
        if (sel == i) dv = c[i];

    if (active) {
        float invn = 1.0f / fmaxf(sqrtf(ss), 1e-12f);
        float sim  = dv * a_inv[aidx_m] * invn * TEMP_INV;
        neg_sim[wbase + m] = sim;
    }
}

// ---------------------------------------------------------------------------
// K3: per-anchor segmented max / log-sum-exp. One wave per anchor; the sorted
// index array gives the segment via binary search (no atomics, deterministic).
// ---------------------------------------------------------------------------
__global__ void k_anchor(const int* __restrict__ idx, const float* __restrict__ neg_sim,
                         const float* __restrict__ pos_sim,
                         float* __restrict__ loss_i, float* __restrict__ corr,
                         int B, int N) {
    int wave = threadIdx.x >> 5, lane = threadIdx.x & 31;
    int b = blockIdx.x * 4 + wave;
    if (b >= B) return;
    // lower_bound(b) and lower_bound(b+1) over sorted idx (uniform per wave)
    int lo = 0, hi = N;
    while (lo < hi) { int mid = (lo + hi) >> 1; if (idx[mid] < b) lo = mid + 1; else hi = mid; }
    int lo2 = lo, hi2 = N;
    while (lo2 < hi2) { int mid = (lo2 + hi2) >> 1; if (idx[mid] <= b) lo2 = mid + 1; else hi2 = mid; }
    int start = lo, end = lo2;

    float pos  = pos_sim[b];
    float smax = -INFINITY;
    for (int j = start + lane; j < end; j += 32) smax = fmaxf(smax, neg_sim[j]);
    smax = wave_reduce_max(smax);

    float li = 0.0f, cr = 0.0f;
    if (end > start) {
        float mm = fmaxf(pos, smax);
        float s  = 0.0f;
        for (int j = start + lane; j < end; j += 32) s += expf(neg_sim[j] - mm);
        s = wave_reduce_add(s);
        s += expf(pos - mm);
        li = -pos + mm + logf(s);
        cr = (pos > smax) ? 1.0f : 0.0f;
    }
    if (lane == 0) { loss_i[b] = li; corr[b] = cr; }
}

// ---------------------------------------------------------------------------
// K4: deterministic tree reduction of loss_i / correct -> d_out[0..1]
// ---------------------------------------------------------------------------
__global__ void k_final(const float* __restrict__ loss_i, const float* __restrict__ corr,
                        float* __restrict__ out, int B) {
    __shared__ float sl[256], sc[256];
    int t = threadIdx.x;
    float a = 0.f, c = 0.f;
    for (int i = t; i < B; i += 256) { a += loss_i[i]; c += corr[i]; }
    sl[t] = a; sc[t] = c;
    __syncthreads();
    for (int o = 128; o > 0; o >>= 1) {
        if (t < o) { sl[t] += sl[t + o]; sc[t] += sc[t + o]; }
        __syncthreads();
    }
    if (t == 0) { out[0] = sl[0] / (float)B; out[1] = sc[0] / (float)B; }
}

// ---------------------------------------------------------------------------
extern "C" void kernel_launch(void* const* d_in, const int* in_sizes, int n_in,
                              void* d_out, int out_size, void* d_ws, size_t ws_size,
                              hipStream_t stream) {
    (void)n_in; (void)out_size; (void)ws_size;
    const float* anchor    = (const float*)d_in[0];
    const float* positive  = (const float*)d_in[1];
    const float* negatives = (const float*)d_in[2];
    const int*   idx       = (const int*)d_in[3];
    int B = in_sizes[0] / D;   // 4096
    int N = in_sizes[3];       // 131072

    float* ws      = (float*)d_ws;
    float* pos_sim = ws;
    float* a_inv   = ws + B;
    float* loss_i  = ws + 2 * B;
    float* corr    = ws + 3 * B;
    float* neg_sim = ws + 4 * B;   // N floats

    k_pos   <<<dim3((B + 3) / 4), dim3(128), 0, stream>>>(anchor, positive, pos_sim, a_inv, B);
    k_neg   <<<dim3(N / (NW * TN)), dim3(128), 0, stream>>>(anchor, negatives, idx, a_inv, neg_sim, N);
    k_anchor<<<dim3((B + 3) / 4), dim3(128), 0, stream>>>(idx, neg_sim, pos_sim, loss_i, corr, B, N);
    k_final <<<dim3(1), dim3(256), 0, stream>>>(loss_i, corr, (float*)d_out, B);
}